// FenwickTree_35270271434816
// MI455X (gfx1250) — compile-verified
//
#include <hip/hip_runtime.h>
#include <math.h>

// ---------------------------------------------------------------------------
// Fenwick TreeLSTM for gfx1250 using V_WMMA_F32_16X16X4_F32 (fp32 WMMA).
//  z = [A(16x512)] @ [W(512x1280)] + b   per 16-row block, then LSTM gates.
// ---------------------------------------------------------------------------

typedef __attribute__((ext_vector_type(2))) float v2f;
typedef __attribute__((ext_vector_type(8))) float v8f;

#define NROWS   32768
#define DDIM    256
#define KDIM    512
#define ZCOLS   1280     // 5 gates * 256
#define KB      128      // 512 / 4 K-steps
#define LDS_STR 516      // padded LDS row stride (floats): 516 % 64 = 4 -> conflict-free b64

__device__ __forceinline__ float sigf(float x) {
    return 1.0f / (1.0f + __expf(-x));
}

// Pack W (512 x 1280 row-major) into WMMA B-fragment order:
//   Wp[(((kb*2 + p)*1280) + c)*2 + q] = W[(kb*4 + 2p + q)*1280 + c]
// Lane (col=c, half=p) then loads its (K, K+1) pair as one coalesced b64.
__global__ __launch_bounds__(256) void pack_w_kernel(const float* __restrict__ W,
                                                     float* __restrict__ Wp) {
    int flat = blockIdx.x * 256 + threadIdx.x;
    if (flat >= KDIM * ZCOLS) return;
    int q  = flat & 1;
    int r  = flat >> 1;
    int c  = r % ZCOLS;
    int s  = r / ZCOLS;
    int p  = s & 1;
    int kb = s >> 1;
    Wp[flat] = W[(size_t)(kb * 4 + 2 * p + q) * ZCOLS + c];
}

// ---------------- Phase 1: one merge level ----------------
// A (M x 512) == prevH viewed contiguously (rows 2i,2i+1 concatenated).
__global__ __launch_bounds__(256) void level_kernel(
    const float* __restrict__ prevH, const float* __restrict__ prevC,
    const float* __restrict__ Wp, const float* __restrict__ bvec,
    float* __restrict__ outH, float* __restrict__ outC, int M) {

    const int tid  = threadIdx.x;
    const int lane = tid & 31;
    const int wave = tid >> 5;
    const int cl16 = lane & 15;
    const int half = lane >> 4;
    const int blockRow = blockIdx.x * 16;

    v8f acc[5][2];
#pragma unroll
    for (int g = 0; g < 5; ++g)
#pragma unroll
        for (int s = 0; s < 2; ++s) {
            float bv = bvec[g * 256 + wave * 32 + s * 16 + cl16];
            v8f t;
#pragma unroll
            for (int e = 0; e < 8; ++e) t[e] = bv;
            acc[g][s] = t;
        }

    // A fragment base: row = cl16, K-phase = half*2
    const float* Arow = prevH + (size_t)(blockRow + cl16) * KDIM + half * 2;

    for (int kb = 0; kb < KB; ++kb) {
        v2f a = *(const v2f*)(Arow + kb * 4);
        const float* wrow = Wp + ((size_t)(kb * 2 + half) * ZCOLS + wave * 32 + cl16) * 2;
#pragma unroll
        for (int g = 0; g < 5; ++g)
#pragma unroll
            for (int s = 0; s < 2; ++s) {
                v2f b = *(const v2f*)(wrow + (g * 256 + s * 16) * 2);
                acc[g][s] = __builtin_amdgcn_wmma_f32_16x16x4_f32(
                    false, a, false, b, (short)0, acc[g][s], false, false);
            }
    }

    // LSTM cell elementwise: c = sig(i)tanh(u) + sig(fl)*cl + sig(fr)*cr ; h = sig(o)tanh(c)
#pragma unroll
    for (int s = 0; s < 2; ++s) {
        int j = wave * 32 + s * 16 + cl16;
#pragma unroll
        for (int v = 0; v < 8; ++v) {
            int row = blockRow + v + 8 * half;
            if (row < M) {
                float cl = prevC[(size_t)(2 * row) * DDIM + j];
                float cr = prevC[(size_t)(2 * row + 1) * DDIM + j];
                float zi = acc[0][s][v], zo = acc[1][s][v], zu = acc[2][s][v];
                float zfl = acc[3][s][v], zfr = acc[4][s][v];
                float c = sigf(zi) * tanhf(zu) + sigf(zfl) * cl + sigf(zfr) * cr;
                float h = sigf(zo) * tanhf(c);
                outH[(size_t)row * DDIM + j] = h;
                outC[(size_t)row * DDIM + j] = c;
            }
        }
    }
}

// ---------------- Phase 2: one prefix-combine iteration ----------------
// state = d_out, (N x 512) rows laid out [h(256) | c(256)], updated in place.
__global__ __launch_bounds__(256) void sum_kernel(
    float* __restrict__ state,
    const float* __restrict__ nodeH, const float* __restrict__ nodeC,
    const float* __restrict__ Wp, const float* __restrict__ bvec,
    int k, int Mk) {

    __shared__ float As[16 * LDS_STR];

    const int tid  = threadIdx.x;
    const int lane = tid & 31;
    const int wave = tid >> 5;
    const int cl16 = lane & 15;
    const int half = lane >> 4;
    const int blockRow = blockIdx.x * 16;

    // Block-uniform early exit: if no row t in [blockRow+1, blockRow+16] has
    // bit k set, this block stores nothing -> skip staging + GEMM entirely.
    // (Uniform across the workgroup: safe w.r.t. __syncthreads and WMMA EXEC.)
    if (k >= 5) {
        int t0 = blockRow + 1;
        int t1 = blockRow + 16;
        int hi0 = t0 >> k;
        int hi1 = t1 >> k;
        if (hi0 == hi1 && ((hi0 & 1) == 0)) return;
    }

    // Stage A = [state_h | gathered node_h] : 16 rows x 512 into LDS
#pragma unroll
    for (int it = 0; it < 32; ++it) {
        int idx = it * 256 + tid;
        int r   = idx >> 9;
        int col = idx & 511;
        int grow = blockRow + r;
        float v;
        if (col < 256) {
            v = state[(size_t)grow * 512 + col];
        } else {
            int t = grow + 1;
            int nidx = (t >> (k + 1)) * 2;
            if (nidx > Mk - 1) nidx = Mk - 1;
            v = nodeH[(size_t)nidx * DDIM + (col - 256)];
        }
        As[r * LDS_STR + col] = v;
    }
    __syncthreads();

    v8f acc[5][2];
#pragma unroll
    for (int g = 0; g < 5; ++g)
#pragma unroll
        for (int s = 0; s < 2; ++s) {
            float bv = bvec[g * 256 + wave * 32 + s * 16 + cl16];
            v8f t;
#pragma unroll
            for (int e = 0; e < 8; ++e) t[e] = bv;
            acc[g][s] = t;
        }

    const float* Arow = &As[cl16 * LDS_STR + half * 2];

    for (int kb = 0; kb < KB; ++kb) {
        v2f a = *(const v2f*)(Arow + kb * 4);
        const float* wrow = Wp + ((size_t)(kb * 2 + half) * ZCOLS + wave * 32 + cl16) * 2;
#pragma unroll
        for (int g = 0; g < 5; ++g)
#pragma unroll
            for (int s = 0; s < 2; ++s) {
                v2f b = *(const v2f*)(wrow + (g * 256 + s * 16) * 2);
                acc[g][s] = __builtin_amdgcn_wmma_f32_16x16x4_f32(
                    false, a, false, b, (short)0, acc[g][s], false, false);
            }
    }

    // masked writeback:  state = has ? (valid ? cell : node) : state
#pragma unroll
    for (int s = 0; s < 2; ++s) {
        int j = wave * 32 + s * 16 + cl16;
#pragma unroll
        for (int v = 0; v < 8; ++v) {
            int grow = blockRow + v + 8 * half;
            int t = grow + 1;
            bool has   = ((t >> k) & 1) != 0;
            bool valid = (t & ((1 << k) - 1)) != 0;
            if (has) {
                int nidx = (t >> (k + 1)) * 2;
                if (nidx > Mk - 1) nidx = Mk - 1;
                float nhv = nodeH[(size_t)nidx * DDIM + j];
                float ncv = nodeC[(size_t)nidx * DDIM + j];
                float hout, cout;
                if (valid) {
                    float clv = state[(size_t)grow * 512 + 256 + j];
                    float zi = acc[0][s][v], zo = acc[1][s][v], zu = acc[2][s][v];
                    float zfl = acc[3][s][v], zfr = acc[4][s][v];
                    float cc = sigf(zi) * tanhf(zu) + sigf(zfl) * clv + sigf(zfr) * ncv;
                    hout = sigf(zo) * tanhf(cc);
                    cout = cc;
                } else {
                    hout = nhv;
                    cout = ncv;
                }
                state[(size_t)grow * 512 + j] = hout;
                state[(size_t)grow * 512 + 256 + j] = cout;
            }
        }
    }
}

// ---------------------------------------------------------------------------
extern "C" void kernel_launch(void* const* d_in, const int* in_sizes, int n_in,
                              void* d_out, int out_size, void* d_ws, size_t ws_size,
                              hipStream_t stream) {
    const float* h_bot   = (const float*)d_in[0];
    const float* c_bot   = (const float*)d_in[1];
    const float* W_merge = (const float*)d_in[2];
    const float* b_merge = (const float*)d_in[3];
    const float* W_sum   = (const float*)d_in[4];
    const float* b_sum   = (const float*)d_in[5];
    float* out = (float*)d_out;
    float* ws  = (float*)d_ws;

    const int L = 15;

    // Workspace layout (floats):
    //   [ packed W_merge : 512*1280 ][ packed W_sum : 512*1280 ]
    //   [ levels_h k=1..15, rows padded to >=16 ][ levels_c k=1..15, same ]
    float* WpM  = ws;
    float* WpS  = ws + (size_t)KDIM * ZCOLS;
    float* base = WpS + (size_t)KDIM * ZCOLS;

    size_t padRows[16];
    for (int k = 1; k <= L; ++k) {
        size_t Mk = (size_t)NROWS >> k;
        padRows[k] = (Mk < 16) ? 16 : Mk;
    }
    float* levH[16];
    float* levC[16];
    size_t cur = 0;
    for (int k = 1; k <= L; ++k) { levH[k] = base + cur; cur += padRows[k] * DDIM; }
    for (int k = 1; k <= L; ++k) { levC[k] = base + cur; cur += padRows[k] * DDIM; }

    // 1) pack both weight matrices into WMMA fragment order
    int packN = KDIM * ZCOLS;
    int packBlocks = (packN + 255) / 256;
    pack_w_kernel<<<packBlocks, 256, 0, stream>>>(W_merge, WpM);
    pack_w_kernel<<<packBlocks, 256, 0, stream>>>(W_sum, WpS);

    // 2) phase 1: build merge-tree levels
    const float* ph = h_bot;
    const float* pc = c_bot;
    for (int k = 1; k <= L; ++k) {
        int Mk = NROWS >> k;
        int blocks = (Mk + 15) / 16;
        level_kernel<<<blocks, 256, 0, stream>>>(ph, pc, WpM, b_merge, levH[k], levC[k], Mk);
        ph = levH[k];
        pc = levC[k];
    }

    // 3) phase 2: Fenwick prefix combine, state lives in d_out ([h|c] per row)
    for (int k = 0; k <= L; ++k) {
        const float* nH = (k == 0) ? h_bot : levH[k];
        const float* nC = (k == 0) ? c_bot : levC[k];
        sum_kernel<<<NROWS / 16, 256, 0, stream>>>(out, nH, nC, WpS, b_sum, k, NROWS >> k);
    }
}